// FraudGNN_17532056502368
// MI455X (gfx1250) — compile-verified
//
#include <hip/hip_runtime.h>

// ---------------------------------------------------------------------------
// FraudGNN: 2-layer GraphSAGE (mean aggr) on MI455X / gfx1250
//   N = 100000 nodes, E = 1.6M edges, d_in = d_hid = 64, d_out = 2
// Roofline: edge scatter is ~1.6 GB of traffic -> memory bound (~70-100us
// at 23.3 TB/s). GEMMs (1.6 GFLOP) done with V_WMMA_F32_16X16X4_F32 so the
// dense math rides the matrix pipe at full fp32 precision.
// ---------------------------------------------------------------------------

#define N_NODES 100000
#define DIM     64
#define TILES_M (N_NODES / 16)   // 6250

typedef __attribute__((ext_vector_type(2))) float v2f;
typedef __attribute__((ext_vector_type(8))) float v8f;

// ---------------------------------------------------------------------------
// Zero workspace accumulators (float4 stores -> global_store_b128)
// ---------------------------------------------------------------------------
__global__ __launch_bounds__(256)
void fg_zero_f4(float4* __restrict__ p, long long n4) {
    long long i = (long long)blockIdx.x * blockDim.x + threadIdx.x;
    if (i < n4) p[i] = make_float4(0.f, 0.f, 0.f, 0.f);
}

// ---------------------------------------------------------------------------
// Edge scatter-add: 64 threads per edge, one feature per thread.
// feat[src[e]] row read is coalesced; accumulation via global_atomic_add_f32
// (non-returning, STOREcnt path). cnt accumulated only on the first pass.
// ---------------------------------------------------------------------------
__global__ __launch_bounds__(256)
void fg_scatter(const float* __restrict__ feat,
                const long long* __restrict__ src,
                const long long* __restrict__ dst,
                float* __restrict__ agg,
                float* __restrict__ cnt,     // nullptr on second pass
                int nEdges) {
    long long gid = (long long)blockIdx.x * blockDim.x + threadIdx.x;
    int e = (int)(gid >> 6);
    int f = (int)(gid & 63);
    if (e >= nEdges) return;
    int s = (int)src[e];
    int d = (int)dst[e];
    unsafeAtomicAdd(agg + (size_t)d * DIM + f, feat[(size_t)s * DIM + f]);
    if (cnt != nullptr && f == 0) {
        unsafeAtomicAdd(cnt + d, 1.0f);
    }
}

// ---------------------------------------------------------------------------
// Layer 1 with WMMA: h = relu( (agg/cnt) @ W1l^T + b1l + x @ W1r^T )
// One wave per 16x16 output tile. 8 waves/block: 2 M-tiles x 4 N-tiles.
// A fragment (16x4 f32): lane L -> row M = L%16, K pair {2*half, 2*half+1}
// B fragment (4x16 f32): lane L -> col N = L%16, same K pairing; B[k][n] =
//   W[n][k] so each lane reads 2 consecutive floats of weight row n.
// C/D (16x16 f32, 8 VGPRs): row = i + 8*half, col = L%16.
// Mean scaling folded into A load; bias + ReLU folded into epilogue.
// ---------------------------------------------------------------------------
__global__ __launch_bounds__(256)
void fg_layer1_wmma(const float* __restrict__ x,
                    const float* __restrict__ agg,
                    const float* __restrict__ cnt,
                    const float* __restrict__ W1l,
                    const float* __restrict__ b1l,
                    const float* __restrict__ W1r,
                    float* __restrict__ h) {
    const int wave = threadIdx.x >> 5;
    const int lane = threadIdx.x & 31;
    const int half = lane >> 4;
    const int nl   = lane & 15;

    const int tileM = blockIdx.x * 2 + (wave >> 2);  // 0..6249
    const int tileN = wave & 3;                      // 0..3

    const int rowA = tileM * 16 + nl;   // A-matrix row this lane feeds
    const int colN = tileN * 16 + nl;   // output column this lane feeds

    const float rcnt = 1.0f / fmaxf(cnt[rowA], 1.0f);

    const float* __restrict__ aggRow = agg + (size_t)rowA * DIM;
    const float* __restrict__ xRow   = x   + (size_t)rowA * DIM;
    const float* __restrict__ wlRow  = W1l + (size_t)colN * DIM;  // B[k][n]=W1l[n][k]
    const float* __restrict__ wrRow  = W1r + (size_t)colN * DIM;

    v8f acc = {};
#pragma unroll
    for (int k0 = 0; k0 < DIM; k0 += 4) {
        const int ko = k0 + 2 * half;
        // mean-branch: A = agg/cnt, B = W1l^T
        v2f a1; a1.x = aggRow[ko] * rcnt; a1.y = aggRow[ko + 1] * rcnt;
        v2f b1; b1.x = wlRow[ko];         b1.y = wlRow[ko + 1];
        acc = __builtin_amdgcn_wmma_f32_16x16x4_f32(false, a1, false, b1,
                                                    (short)0, acc, false, false);
        // root-branch: A = x, B = W1r^T
        v2f a2; a2.x = xRow[ko];  a2.y = xRow[ko + 1];
        v2f b2; b2.x = wrRow[ko]; b2.y = wrRow[ko + 1];
        acc = __builtin_amdgcn_wmma_f32_16x16x4_f32(false, a2, false, b2,
                                                    (short)0, acc, false, false);
    }

    const float bias = b1l[colN];
#pragma unroll
    for (int i = 0; i < 8; ++i) {
        const int row = tileM * 16 + i + 8 * half;
        const float v = acc[i] + bias;
        h[(size_t)row * DIM + colN] = fmaxf(v, 0.0f);
    }
}

// ---------------------------------------------------------------------------
// Layer 2: out[m, 0:2] = (agg2[m]/cnt[m]) @ W2l^T + b2l + h[m] @ W2r^T
// d_out = 2 -> wave-per-row VALU dot products + shfl_xor tree reduction.
// ---------------------------------------------------------------------------
__global__ __launch_bounds__(256)
void fg_layer2(const float* __restrict__ agg2,
               const float* __restrict__ cnt,
               const float* __restrict__ h,
               const float* __restrict__ W2l,
               const float* __restrict__ b2l,
               const float* __restrict__ W2r,
               float* __restrict__ out) {
    const int row  = blockIdx.x * 8 + (threadIdx.x >> 5);
    const int lane = threadIdx.x & 31;
    if (row >= N_NODES) return;   // whole wave exits together (uniform row)

    const float rcnt = 1.0f / fmaxf(cnt[row], 1.0f);
    const float* __restrict__ aRow = agg2 + (size_t)row * DIM;
    const float* __restrict__ hRow = h    + (size_t)row * DIM;

    float p0 = 0.f, p1 = 0.f;
#pragma unroll
    for (int kk = 0; kk < 2; ++kk) {
        const int k  = lane + kk * 32;
        const float m  = aRow[k] * rcnt;
        const float hv = hRow[k];
        p0 += m * W2l[k]       + hv * W2r[k];
        p1 += m * W2l[DIM + k] + hv * W2r[DIM + k];
    }
#pragma unroll
    for (int off = 16; off >= 1; off >>= 1) {
        p0 += __shfl_xor(p0, off, 32);
        p1 += __shfl_xor(p1, off, 32);
    }
    if (lane == 0) {
        out[(size_t)row * 2 + 0] = p0 + b2l[0];
        out[(size_t)row * 2 + 1] = p1 + b2l[1];
    }
}

// ---------------------------------------------------------------------------
// Launch: zero -> scatter(x)+cnt -> WMMA layer1 -> scatter(h) -> layer2
// ---------------------------------------------------------------------------
extern "C" void kernel_launch(void* const* d_in, const int* in_sizes, int n_in,
                              void* d_out, int out_size, void* d_ws, size_t ws_size,
                              hipStream_t stream) {
    const float*     x   = (const float*)d_in[0];
    const long long* ei  = (const long long*)d_in[1];   // int64 [2, E]
    const float*     W1l = (const float*)d_in[2];
    const float*     b1l = (const float*)d_in[3];
    const float*     W1r = (const float*)d_in[4];
    const float*     W2l = (const float*)d_in[5];
    const float*     b2l = (const float*)d_in[6];
    const float*     W2r = (const float*)d_in[7];
    float*           out = (float*)d_out;

    const int E = in_sizes[1] / 2;
    const long long* src = ei;
    const long long* dst = ei + E;

    // Workspace layout (floats):
    //   agg1 [N*64] | agg2 [N*64] | cnt [N] | h [N*64]
    float* ws   = (float*)d_ws;
    float* agg1 = ws;
    float* agg2 = ws + (size_t)N_NODES * DIM;
    float* cnt  = ws + (size_t)2 * N_NODES * DIM;
    float* h    = ws + (size_t)2 * N_NODES * DIM + N_NODES;

    // 1) zero agg1, agg2, cnt (contiguous: 2*N*64 + N floats, /4 for float4)
    {
        const long long n4 = ((long long)2 * N_NODES * DIM + N_NODES) / 4;
        const int blocks = (int)((n4 + 255) / 256);
        fg_zero_f4<<<blocks, 256, 0, stream>>>((float4*)ws, n4);
    }
    // 2) scatter x -> agg1 (+ degree counts)
    {
        const long long threads = (long long)E * DIM;
        const int blocks = (int)((threads + 255) / 256);
        fg_scatter<<<blocks, 256, 0, stream>>>(x, src, dst, agg1, cnt, E);
    }
    // 3) layer 1 (WMMA f32 16x16x4): h = relu(mean@W1l^T + b1l + x@W1r^T)
    {
        fg_layer1_wmma<<<TILES_M / 2, 256, 0, stream>>>(x, agg1, cnt,
                                                        W1l, b1l, W1r, h);
    }
    // 4) scatter h -> agg2 (reuse cnt: degrees depend only on edges)
    {
        const long long threads = (long long)E * DIM;
        const int blocks = (int)((threads + 255) / 256);
        fg_scatter<<<blocks, 256, 0, stream>>>(h, src, dst, agg2, nullptr, E);
    }
    // 5) layer 2: out = mean2@W2l^T + b2l + h@W2r^T
    {
        const int blocks = (N_NODES + 7) / 8;
        fg_layer2<<<blocks, 256, 0, stream>>>(agg2, cnt, h, W2l, b2l, W2r, out);
    }
}